// GNNEncoder_84000970375718
// MI455X (gfx1250) — compile-verified
//
#include <hip/hip_runtime.h>
#include <hip/hip_bf16.h>

typedef __attribute__((ext_vector_type(2))) float v2f;
typedef __attribute__((ext_vector_type(8))) float v8f;

#define HID 128

// ---------------------------------------------------------------------------
// WMMA wrapper: the builtin only exists in the device compilation pass, so
// guard with __gfx1250__ and give the host pass a parse-only placeholder.
// ---------------------------------------------------------------------------
__device__ inline v8f wmma_f32_16x16x4(v2f a, v2f b, v8f c) {
#if defined(__gfx1250__)
    return __builtin_amdgcn_wmma_f32_16x16x4_f32(false, a, false, b, (short)0, c,
                                                 false, false);
#else
    return c;  // host pass placeholder; never executed
#endif
}

// ---------------------------------------------------------------------------
// Degree / normalization
// ---------------------------------------------------------------------------
__global__ void k_deg_init(float* __restrict__ deg, int n) {
    int i = blockIdx.x * blockDim.x + threadIdx.x;
    if (i < n) deg[i] = 1.0f;  // self-loop contribution
}

__global__ void k_deg_accum(const int* __restrict__ ei, float* __restrict__ deg, int E) {
    int e = blockIdx.x * blockDim.x + threadIdx.x;
    if (e < E) atomicAdd(&deg[ei[E + e]], 1.0f);  // dst row of edge_index
}

__global__ void k_rsqrt_inplace(float* __restrict__ deg, int n) {
    int i = blockIdx.x * blockDim.x + threadIdx.x;
    if (i < n) deg[i] = rsqrtf(fmaxf(deg[i], 1.0f));
}

// ---------------------------------------------------------------------------
// Layer 1 GEMM: h1 = x @ W1   (K = 6, VALU is the right tool)
// ---------------------------------------------------------------------------
__global__ void k_gemm1(const float* __restrict__ x, const float* __restrict__ W1,
                        float* __restrict__ h, int n) {
    __shared__ float w[6 * HID];
    for (int i = threadIdx.x; i < 6 * HID; i += blockDim.x) w[i] = W1[i];
    __syncthreads();
    int idx = blockIdx.x * blockDim.x + threadIdx.x;  // over n*128
    int i = idx >> 7, f = idx & 127;
    if (i < n) {
        const float* xr = x + (size_t)i * 6;
        float s = 0.f;
#pragma unroll
        for (int k = 0; k < 6; ++k) s = fmaf(xr[k], w[k * HID + f], s);
        h[idx] = s;
    }
}

// ---------------------------------------------------------------------------
// acc = b + dis[i]^2 * h[i]   (self-loop term; also serves as initialization)
// ---------------------------------------------------------------------------
__global__ void k_init_acc(const float* __restrict__ h, const float* __restrict__ dis,
                           const float* __restrict__ b, float* __restrict__ acc, int n) {
    int idx = blockIdx.x * blockDim.x + threadIdx.x;
    int i = idx >> 7, f = idx & 127;
    if (i < n) {
        float d = dis[i];
        acc[idx] = fmaf(d * d, h[idx], b[f]);
    }
}

// ---------------------------------------------------------------------------
// Edge scatter: acc[dst] += (dis[src]*dis[dst]) * h[src]
// one wave (32 lanes) per edge, float4 per lane => coalesced 512B gather
// ---------------------------------------------------------------------------
__global__ void k_scatter(const int* __restrict__ ei, const float* __restrict__ dis,
                          const float* __restrict__ h, float* __restrict__ acc, int E) {
    int e = blockIdx.x * 8 + (threadIdx.x >> 5);
    int lane = threadIdx.x & 31;
    if (e < E) {
        int s = ei[e];
        int d = ei[E + e];
        float nrm = dis[s] * dis[d];
        float4 v = ((const float4*)(h + (size_t)s * HID))[lane];
        float* ap = acc + (size_t)d * HID + lane * 4;
        atomicAdd(ap + 0, v.x * nrm);
        atomicAdd(ap + 1, v.y * nrm);
        atomicAdd(ap + 2, v.z * nrm);
        atomicAdd(ap + 3, v.w * nrm);
    }
}

__global__ void k_relu(float* __restrict__ a, int total) {
    int idx = blockIdx.x * blockDim.x + threadIdx.x;
    if (idx < total) a[idx] = fmaxf(a[idx], 0.f);
}

// ---------------------------------------------------------------------------
// Layer 2 GEMM: out = A(n x 128) @ W(128 x 128) using V_WMMA_F32_16X16X4_F32.
// Block = 256 threads = 8 waves; block tile = 128 rows.
// Wave w computes rows [w*16, w*16+16) x all 128 cols (8 column tiles).
// K is staged through LDS in two halves of 64 -> 64 KB LDS total.
// Fragment layouts per CDNA5 ISA 7.12.2 (f32 16x16x4):
//   A: lane L<16 holds (M=L, K=k0,k0+1); lane L>=16 holds (M=L-16, K=k0+2,k0+3)
//   B: VGPR0 = row k0 (lanes 0-15) / row k0+2 (lanes 16-31), VGPR1 = +1
//   C: VGPR r = row r (lanes 0-15) / row r+8 (lanes 16-31)
// ---------------------------------------------------------------------------
__global__ void k_gemm2_wmma(const float* __restrict__ A, const float* __restrict__ W,
                             float* __restrict__ out, int n) {
    __shared__ float lA[128 * 64];  // 128 rows x 64 K
    __shared__ float lB[64 * 128];  // 64 K rows x 128 cols
    const int tid = threadIdx.x;
    const int rowBase = blockIdx.x * 128;
    const int wave = tid >> 5;
    const int lane = tid & 31;
    const int half = lane >> 4;  // 0: lanes 0-15, 1: lanes 16-31
    const int l15 = lane & 15;
    const int arow = wave * 16 + l15;  // M row within block tile for this lane

    v8f acc[8];
#pragma unroll
    for (int nt = 0; nt < 8; ++nt) acc[nt] = 0.f;

    for (int kb = 0; kb < 2; ++kb) {
        __syncthreads();
        // stage A rows (bounds-checked), 64-wide K slice
        for (int i = tid; i < 128 * 16; i += 256) {  // float4 count
            int r = i >> 4, c4 = i & 15;
            int gr = rowBase + r;
            float4 v = make_float4(0.f, 0.f, 0.f, 0.f);
            if (gr < n) v = ((const float4*)A)[(size_t)gr * 32 + kb * 16 + c4];
            ((float4*)lA)[i] = v;
        }
        // stage W K-slice (rows kb*64 .. kb*64+63, all 128 cols)
        for (int i = tid; i < 64 * 32; i += 256) {  // float4 count
            int r = i >> 5, c4 = i & 31;
            ((float4*)lB)[i] = ((const float4*)W)[(size_t)(kb * 64 + r) * 32 + c4];
        }
        __syncthreads();

#pragma unroll
        for (int kk = 0; kk < 16; ++kk) {
            int k0 = kk * 4 + 2 * half;  // this lane's K pair
            v2f a;
            a.x = lA[arow * 64 + k0];
            a.y = lA[arow * 64 + k0 + 1];
#pragma unroll
            for (int nt = 0; nt < 8; ++nt) {
                v2f b;
                b.x = lB[k0 * 128 + nt * 16 + l15];
                b.y = lB[(k0 + 1) * 128 + nt * 16 + l15];
                acc[nt] = wmma_f32_16x16x4(a, b, acc[nt]);
            }
        }
    }

    // store D: VGPR r -> row (r + 8*half), col nt*16 + l15
#pragma unroll
    for (int nt = 0; nt < 8; ++nt) {
#pragma unroll
        for (int r = 0; r < 8; ++r) {
            int grow = rowBase + wave * 16 + r + 8 * half;
            if (grow < n) out[(size_t)grow * HID + nt * 16 + l15] = acc[nt][r];
        }
    }
}

// ---------------------------------------------------------------------------
// Pooling
// ---------------------------------------------------------------------------
__global__ void k_zero(float* __restrict__ p, int total) {
    int i = blockIdx.x * blockDim.x + threadIdx.x;
    if (i < total) p[i] = 0.f;
}

__global__ void k_counts(const int* __restrict__ batch, float* __restrict__ counts, int n) {
    int i = blockIdx.x * blockDim.x + threadIdx.x;
    if (i < n) atomicAdd(&counts[batch[i]], 1.0f);
}

__global__ void k_pool_accum(const int* __restrict__ batch, const float* __restrict__ acc,
                             float* __restrict__ out, int n) {
    int idx = blockIdx.x * blockDim.x + threadIdx.x;
    int i = idx >> 7, f = idx & 127;
    if (i < n) atomicAdd(&out[(size_t)batch[i] * HID + f], acc[idx]);
}

__global__ void k_pool_div(float* __restrict__ out, const float* __restrict__ counts, int total) {
    int idx = blockIdx.x * blockDim.x + threadIdx.x;
    if (idx < total) out[idx] /= fmaxf(counts[idx >> 7], 1.0f);
}

// ---------------------------------------------------------------------------
// Launch
// ---------------------------------------------------------------------------
extern "C" void kernel_launch(void* const* d_in, const int* in_sizes, int n_in,
                              void* d_out, int out_size, void* d_ws, size_t ws_size,
                              hipStream_t stream) {
    const float* x     = (const float*)d_in[0];
    const int*   ei    = (const int*)d_in[1];
    const int*   batch = (const int*)d_in[2];
    const float* W1    = (const float*)d_in[4];
    const float* b1    = (const float*)d_in[5];
    const float* W2    = (const float*)d_in[6];
    const float* b2    = (const float*)d_in[7];
    float* out = (float*)d_out;

    const int N = in_sizes[0] / 6;
    const int E = in_sizes[1] / 2;
    const int G = out_size / HID;

    // workspace layout (floats)
    float* ws = (float*)d_ws;
    float* dis    = ws;                       // N
    float* counts = ws + N;                   // G
    size_t ofs = ((size_t)(N + G) + 127) & ~(size_t)127;
    float* bufH = ws + ofs;                   // N*128 : h1 / h2
    float* bufA = bufH + (size_t)N * HID;     // N*128 : acc1(=relu h) / acc2

    const int T = 256;
    const int nBlkN   = (N + T - 1) / T;
    const int nBlkE   = (E + T - 1) / T;
    const int nBlkNF  = (int)(((size_t)N * HID + T - 1) / T);
    const int nBlkEW  = (E + 7) / 8;          // one wave per edge
    const int nBlkGm  = (N + 127) / 128;      // 128-row tiles
    const int nBlkOut = (G * HID + T - 1) / T;

    // normalization
    k_deg_init<<<nBlkN, T, 0, stream>>>(dis, N);
    k_deg_accum<<<nBlkE, T, 0, stream>>>(ei, dis, E);
    k_rsqrt_inplace<<<nBlkN, T, 0, stream>>>(dis, N);

    // layer 1
    k_gemm1<<<nBlkNF, T, 0, stream>>>(x, W1, bufH, N);
    k_init_acc<<<nBlkNF, T, 0, stream>>>(bufH, dis, b1, bufA, N);
    k_scatter<<<nBlkEW, T, 0, stream>>>(ei, dis, bufH, bufA, E);
    k_relu<<<nBlkNF, T, 0, stream>>>(bufA, N * HID);

    // layer 2 (WMMA f32): bufH = relu(acc1) @ W2
    k_gemm2_wmma<<<nBlkGm, T, 0, stream>>>(bufA, W2, bufH, N);
    k_init_acc<<<nBlkNF, T, 0, stream>>>(bufH, dis, b2, bufA, N);
    k_scatter<<<nBlkEW, T, 0, stream>>>(ei, dis, bufH, bufA, E);

    // mean pool
    k_zero<<<(G + T - 1) / T, T, 0, stream>>>(counts, G);
    k_counts<<<nBlkN, T, 0, stream>>>(batch, counts, N);
    k_zero<<<nBlkOut, T, 0, stream>>>(out, G * HID);
    k_pool_accum<<<nBlkNF, T, 0, stream>>>(batch, bufA, out, N);
    k_pool_div<<<nBlkOut, T, 0, stream>>>(out, counts, G * HID);
}